// StateTransitionModel_47270410060466
// MI455X (gfx1250) — compile-verified
//
#include <hip/hip_runtime.h>
#include <hip/hip_bf16.h>

// ---------------------------------------------------------------------------
// Mamba (2-layer) forward for MI455X (gfx1250, wave32).
// - GEMMs: V_WMMA_F32_16X16X4_F32, one wave owns NT 16x16 tiles (A-fragment
//   reuse across N), software-pipelined fragment loads (next k-step's loads
//   issued before current k-step's WMMAs so s_wait_loadcnt overlaps).
// - Selective scan: segmented (32 segments x 128 steps) to break the serial
//   chain; B/C staged to LDS with GLOBAL_LOAD_ASYNC_TO_LDS_B128 (ASYNCcnt).
// ---------------------------------------------------------------------------

typedef __attribute__((ext_vector_type(2))) float v2f;
typedef __attribute__((ext_vector_type(8))) float v8f;

#define D_MODEL  256
#define D_INNER  512
#define D_STATE  16
#define D_CONV   4
#define DT_RANK  16
#define BATCH    2
#define SEQ      4096
#define MTOT     (BATCH * SEQ)          // 8192 rows for all GEMMs
#define SEGLEN   128
#define NSEG     (SEQ / SEGLEN)         // 32

__device__ __forceinline__ float softplusf(float x) {
    return (x > 20.f) ? x : log1pf(__expf(x));
}
__device__ __forceinline__ float siluf(float x) {
    return x / (1.f + __expf(-x));
}

// CDNA5 async global->LDS copy (16B per lane), tracked with ASYNCcnt.
__device__ __forceinline__ void async_copy_b128(unsigned lds_byte_addr,
                                                const float* gsrc) {
    asm volatile("global_load_async_to_lds_b128 %0, %1, off"
                 :: "v"(lds_byte_addr), "v"(gsrc)
                 : "memory");
}
__device__ __forceinline__ void async_wait0() {
    asm volatile("s_wait_asynccnt 0x0" ::: "memory");
}

// ---------------------------------------------------------------------------
// fp32 GEMM:  C[M x N] = A[M x K] * W[N x K]^T  (+bias) (+softplus)
// One wave computes NT adjacent 16x16 tiles via V_WMMA_F32_16X16X4_F32,
// reusing a single A-fragment across the NT B-fragments.  Fragment loads are
// software-pipelined one k-step ahead of the WMMAs.
// Block = 256 threads = 8 waves; grid = (N/(16*NT), M/128).
// A 16x4 per-lane: M = lane%16, K-half = lane/16 (2 consecutive K).
// B 4x16 per-lane (symmetric): N = lane%16, K-half = lane/16.
// C/D: VGPR r -> row r + 8*(lane>=16), col = lane%16.
// EXEC all-1s at every WMMA (no divergence), per ISA 7.12.
// ---------------------------------------------------------------------------
template<int NT>
__global__ void gemm_wmma_f32(const float* __restrict__ A,
                              const float* __restrict__ W,
                              float* __restrict__ C,
                              int K, int lda, int ldb, int ldc,
                              const float* __restrict__ bias, int act)
{
    const int lane = threadIdx.x & 31;
    const int wave = threadIdx.x >> 5;
    const int m0   = blockIdx.y * 128 + wave * 16;
    const int n0   = blockIdx.x * (16 * NT);
    const int mr   = m0 + (lane & 15);
    const int kh   = (lane >> 4) << 1;              // 0 or 2

    const float* ap = A + (size_t)mr * lda + kh;
    // Independent per-tile B pointers (no serial u64 add chain in the loop).
    const float* bpt[NT];
#pragma unroll
    for (int t = 0; t < NT; ++t)
        bpt[t] = W + (size_t)(n0 + t * 16 + (lane & 15)) * ldb + kh;

    __builtin_prefetch(ap, 0, 3);                   // global_prefetch_b8
    __builtin_prefetch(bpt[0], 0, 3);

    const v8f zero = {0.f, 0.f, 0.f, 0.f, 0.f, 0.f, 0.f, 0.f};
    v8f acc[NT];
#pragma unroll
    for (int t = 0; t < NT; ++t) acc[t] = zero;

    // Prologue: fragments for k = 0.
    v2f av = *(const v2f*)ap;
    v2f bv[NT];
#pragma unroll
    for (int t = 0; t < NT; ++t) bv[t] = *(const v2f*)bpt[t];

    // Pipelined main loop: issue k-step loads before (k-4)-step WMMAs.
    for (int k = 4; k < K; k += 4) {
        const v2f av_n = *(const v2f*)(ap + k);
        v2f bv_n[NT];
#pragma unroll
        for (int t = 0; t < NT; ++t) bv_n[t] = *(const v2f*)(bpt[t] + k);

#pragma unroll
        for (int t = 0; t < NT; ++t)
            acc[t] = __builtin_amdgcn_wmma_f32_16x16x4_f32(
                false, av, false, bv[t], (short)0, acc[t], false, false);

        av = av_n;
#pragma unroll
        for (int t = 0; t < NT; ++t) bv[t] = bv_n[t];
    }
    // Epilogue WMMAs for the final k-step.
#pragma unroll
    for (int t = 0; t < NT; ++t)
        acc[t] = __builtin_amdgcn_wmma_f32_16x16x4_f32(
            false, av, false, bv[t], (short)0, acc[t], false, false);

    const int rbase = m0 + ((lane >> 4) ? 8 : 0);
#pragma unroll
    for (int t = 0; t < NT; ++t) {
        const int nc = n0 + t * 16 + (lane & 15);
        const float bb = bias ? bias[nc] : 0.f;
#pragma unroll
        for (int r = 0; r < 8; ++r) {
            float v = acc[t][r] + bb;
            if (act == 1) v = softplusf(v);
            C[(size_t)(rbase + r) * ldc + nc] = v;
        }
    }
}

// ---------------------------------------------------------------------------
// Causal depthwise conv1d (k=4) + bias + SiLU.  x lives in cols [0,512) of xz.
// ---------------------------------------------------------------------------
__global__ void conv_silu_kernel(const float* __restrict__ xz,
                                 const float* __restrict__ cw,
                                 const float* __restrict__ cb,
                                 float* __restrict__ xc)
{
    int idx = blockIdx.x * blockDim.x + threadIdx.x;
    if (idx >= MTOT * D_INNER) return;
    const int d = idx % D_INNER;
    const int l = (idx / D_INNER) % SEQ;
    const int b = idx / (D_INNER * SEQ);

    const float* base = xz + (size_t)b * SEQ * (2 * D_INNER) + d;
    float acc = cb[d];
#pragma unroll
    for (int j = 0; j < D_CONV; ++j) {
        const int ls = l - (D_CONV - 1) + j;
        if (ls >= 0) acc += base[(size_t)ls * (2 * D_INNER)] * cw[d * D_CONV + j];
    }
    xc[idx] = siluf(acc);
}

// ---------------------------------------------------------------------------
// Segmented selective scan.  h = dA ⊙ h + (dt*x)*B is elementwise linear:
// phase1 = per-segment local (h_local, ∏dA) with h0=0 (parallel over 128 WGs)
// phase2 = serial combine across 32 segments (one thread per (b,d,n))
// phase3 = replay with corrected h0, emit y (+D*x, *silu(gate))
// ---------------------------------------------------------------------------
__global__ void scan_phase1(const float* __restrict__ dtbuf,
                            const float* __restrict__ xdbl,
                            const float* __restrict__ xc,
                            const float* __restrict__ A_log,
                            float* __restrict__ segH,
                            float* __restrict__ segP)
{
    const int d   = blockIdx.x * 256 + threadIdx.x;
    const int seg = blockIdx.y;
    const int b   = blockIdx.z;

    __shared__ float Bs[SEGLEN][D_STATE];
    // Async-stage B[seg] into LDS: 2048 floats, 2 rounds x 256 lanes x 16B.
    {
        const unsigned lds0 = (unsigned)(uintptr_t)(&Bs[0][0]);
#pragma unroll
        for (int rnd = 0; rnd < 2; ++rnd) {
            const int flat = rnd * 1024 + threadIdx.x * 4;
            const int row = flat >> 4, col = flat & 15;
            async_copy_b128(lds0 + flat * 4,
                xdbl + ((size_t)b * SEQ + seg * SEGLEN + row) * 48 + DT_RANK + col);
        }
        async_wait0();
    }
    __syncthreads();

    float Areg[D_STATE], h[D_STATE], p[D_STATE];
#pragma unroll
    for (int n = 0; n < D_STATE; ++n) {
        Areg[n] = -__expf(A_log[d * D_STATE + n]);
        h[n] = 0.f;
        p[n] = 1.f;
    }

    for (int li = 0; li < SEGLEN; ++li) {
        const size_t off = ((size_t)b * SEQ + seg * SEGLEN + li) * D_INNER + d;
        const float dt  = dtbuf[off];
        const float xv  = xc[off];
        const float dtx = dt * xv;
#pragma unroll
        for (int n = 0; n < D_STATE; ++n) {
            const float dA = __expf(dt * Areg[n]);
            h[n] = dA * h[n] + dtx * Bs[li][n];
            p[n] *= dA;
        }
    }

    const size_t so = (((size_t)b * NSEG + seg) * D_INNER + d) * D_STATE;
#pragma unroll
    for (int n = 0; n < D_STATE; ++n) {
        segH[so + n] = h[n];
        segP[so + n] = p[n];
    }
}

__global__ void scan_phase2(float* __restrict__ segH,
                            const float* __restrict__ segP)
{
    const int idx = blockIdx.x * 256 + threadIdx.x;   // over B * D_INNER * D_STATE
    if (idx >= BATCH * D_INNER * D_STATE) return;
    const int n = idx % D_STATE;
    const int d = (idx / D_STATE) % D_INNER;
    const int b = idx / (D_STATE * D_INNER);

    float carry = 0.f;
    for (int seg = 0; seg < NSEG; ++seg) {
        const size_t o = (((size_t)b * NSEG + seg) * D_INNER + d) * D_STATE + n;
        const float hloc = segH[o];
        segH[o] = carry;                    // becomes h0 entering this segment
        carry = segP[o] * carry + hloc;
    }
}

__global__ void scan_phase3(const float* __restrict__ dtbuf,
                            const float* __restrict__ xdbl,
                            const float* __restrict__ xz,
                            const float* __restrict__ A_log,
                            const float* __restrict__ Dp,
                            const float* __restrict__ segH,
                            float* __restrict__ xc)   // in: conv(x); out: y
{
    const int d   = blockIdx.x * 256 + threadIdx.x;
    const int seg = blockIdx.y;
    const int b   = blockIdx.z;

    __shared__ float BC[SEGLEN][2 * D_STATE];
    // Async-stage B|C[seg] into LDS: 4096 floats, 4 rounds x 256 lanes x 16B.
    {
        const unsigned lds0 = (unsigned)(uintptr_t)(&BC[0][0]);
#pragma unroll
        for (int rnd = 0; rnd < 4; ++rnd) {
            const int flat = rnd * 1024 + threadIdx.x * 4;
            const int row = flat >> 5, col = flat & 31;
            async_copy_b128(lds0 + flat * 4,
                xdbl + ((size_t)b * SEQ + seg * SEGLEN + row) * 48 + DT_RANK + col);
        }
        async_wait0();
    }
    __syncthreads();

    float Areg[D_STATE], h[D_STATE];
    const size_t so = (((size_t)b * NSEG + seg) * D_INNER + d) * D_STATE;
#pragma unroll
    for (int n = 0; n < D_STATE; ++n) {
        Areg[n] = -__expf(A_log[d * D_STATE + n]);
        h[n] = segH[so + n];
    }
    const float Dd = Dp[d];

    for (int li = 0; li < SEGLEN; ++li) {
        const size_t row = (size_t)b * SEQ + seg * SEGLEN + li;
        const size_t off = row * D_INNER + d;
        const float dt  = dtbuf[off];
        const float xv  = xc[off];
        const float dtx = dt * xv;
        float y = 0.f;
#pragma unroll
        for (int n = 0; n < D_STATE; ++n) {
            const float dA = __expf(dt * Areg[n]);
            h[n] = dA * h[n] + dtx * BC[li][n];
            y += h[n] * BC[li][D_STATE + n];
        }
        y += Dd * xv;
        const float g = xz[row * (2 * D_INNER) + D_INNER + d];
        y *= siluf(g);
        xc[off] = y;   // overwrite conv buffer in-place with y
    }
}

// ---------------------------------------------------------------------------
// Host-side orchestration (stream-only; graph-capture safe).
// ---------------------------------------------------------------------------
extern "C" void kernel_launch(void* const* d_in, const int* in_sizes, int n_in,
                              void* d_out, int out_size, void* d_ws, size_t ws_size,
                              hipStream_t stream)
{
    const float* z    = (const float*)d_in[0];
    const float* inw  = (const float*)d_in[1];
    const float* cw   = (const float*)d_in[2];
    const float* cb   = (const float*)d_in[3];
    const float* xpw  = (const float*)d_in[4];
    const float* dtw  = (const float*)d_in[5];
    const float* dtb  = (const float*)d_in[6];
    const float* Alog = (const float*)d_in[7];
    const float* Dpar = (const float*)d_in[8];
    const float* outw = (const float*)d_in[9];

    float* ws   = (float*)d_ws;
    float* xz   = ws;                                    // MTOT * 1024
    float* xc   = xz   + (size_t)MTOT * 2 * D_INNER;     // MTOT * 512
    float* xdbl = xc   + (size_t)MTOT * D_INNER;         // MTOT * 48
    float* dt   = xdbl + (size_t)MTOT * 48;              // MTOT * 512
    float* io   = dt   + (size_t)MTOT * D_INNER;         // MTOT * 256
    float* segH = io   + (size_t)MTOT * D_MODEL;         // B*NSEG*D_INNER*D_STATE
    float* segP = segH + (size_t)BATCH * NSEG * D_INNER * D_STATE;

    const float* lin = z;
    for (int layer = 0; layer < 2; ++layer) {
        const float* w_in  = inw  + (size_t)layer * 2 * D_INNER * D_MODEL;
        const float* w_cw  = cw   + (size_t)layer * D_INNER * D_CONV;
        const float* w_cb  = cb   + (size_t)layer * D_INNER;
        const float* w_xp  = xpw  + (size_t)layer * (DT_RANK + 2 * D_STATE) * D_INNER;
        const float* w_dt  = dtw  + (size_t)layer * D_INNER * DT_RANK;
        const float* b_dt  = dtb  + (size_t)layer * D_INNER;
        const float* w_Al  = Alog + (size_t)layer * D_INNER * D_STATE;
        const float* w_D   = Dpar + (size_t)layer * D_INNER;
        const float* w_out = outw + (size_t)layer * D_MODEL * D_INNER;
        float* lout = (layer == 1) ? (float*)d_out : io;

        // 1) xz = lin @ in_w^T    [8192 x 1024], NT=4 -> grid.x = 1024/64
        gemm_wmma_f32<4><<<dim3(16, MTOT / 128), 256, 0, stream>>>(
            lin, w_in, xz, D_MODEL, D_MODEL, D_MODEL, 2 * D_INNER, nullptr, 0);
        // 2) xc = silu(conv1d(x) + b)
        conv_silu_kernel<<<(MTOT * D_INNER + 255) / 256, 256, 0, stream>>>(
            xz, w_cw, w_cb, xc);
        // 3) x_dbl = xc @ xp_w^T  [8192 x 48], NT=3 -> grid.x = 1
        gemm_wmma_f32<3><<<dim3(1, MTOT / 128), 256, 0, stream>>>(
            xc, w_xp, xdbl, D_INNER, D_INNER, D_INNER, 48, nullptr, 0);
        // 4) dt = softplus(x_dbl[:, :16] @ dt_w^T + dt_b)  [8192 x 512], NT=4
        gemm_wmma_f32<4><<<dim3(8, MTOT / 128), 256, 0, stream>>>(
            xdbl, w_dt, dt, DT_RANK, 48, DT_RANK, D_INNER, b_dt, 1);
        // 5) segmented selective scan -> y (in-place into xc)
        scan_phase1<<<dim3(D_INNER / 256, NSEG, BATCH), 256, 0, stream>>>(
            dt, xdbl, xc, w_Al, segH, segP);
        scan_phase2<<<(BATCH * D_INNER * D_STATE + 255) / 256, 256, 0, stream>>>(
            segH, segP);
        scan_phase3<<<dim3(D_INNER / 256, NSEG, BATCH), 256, 0, stream>>>(
            dt, xdbl, xz, w_Al, w_D, segH, xc);
        // 6) out = y @ out_w^T    [8192 x 256], NT=4 -> grid.x = 4
        gemm_wmma_f32<4><<<dim3(4, MTOT / 128), 256, 0, stream>>>(
            xc, w_out, lout, D_INNER, D_INNER, D_INNER, D_MODEL, nullptr, 0);

        lin = lout;
    }
}